// MatchingLoss_27874337751471
// MI455X (gfx1250) — compile-verified
//
#include <hip/hip_runtime.h>
#include <math.h>

typedef float v2f __attribute__((ext_vector_type(2)));
typedef float v8f __attribute__((ext_vector_type(8)));

#define NEG_INF   (-1.0e9f)
#define MARGIN_F  0.2f
#define TEMP_W    0.03f
#define QTILE     128
#define BLOCK     128   // 4 waves; each wave owns 32 q-rows (2 x 16-row subtiles)

// ---------------------------------------------------------------- init
__global__ void k_init(double* acc) {
  if (threadIdx.x < 8) acc[threadIdx.x] = 0.0;
}

// ---------------------------------------------------------------- top-2 NN per (b, r)
// blockIdx.x = b, blockDim.x = R
__global__ void k_nn(const float* __restrict__ coords, const float* __restrict__ ref_mask,
                     int* __restrict__ nn, int R) {
  extern __shared__ float sm[];
  float* cs = sm;           // R*3
  float* ms = sm + R * 3;   // R
  int b = blockIdx.x;
  for (int i = threadIdx.x; i < R * 3; i += blockDim.x) cs[i] = coords[(size_t)b * R * 3 + i];
  for (int i = threadIdx.x; i < R; i += blockDim.x)     ms[i] = ref_mask[(size_t)b * R + i];
  __syncthreads();
  int r = threadIdx.x;
  if (r < R) {
    float x = cs[r * 3 + 0], y = cs[r * 3 + 1], z = cs[r * 3 + 2];
    float d1 = INFINITY, d2 = INFINITY;
    int   i1 = -1,       i2 = -1;
    for (int s = 0; s < R; ++s) {
      if (s == r) continue;
      if (ms[s] < 0.5f) continue;
      float dx = cs[s * 3 + 0] - x;
      float dy = cs[s * 3 + 1] - y;
      float dz = cs[s * 3 + 2] - z;
      float d = dx * dx + dy * dy + dz * dz;
      if (d < d1)      { d2 = d1; i2 = i1; d1 = d; i1 = s; }
      else if (d < d2) { d2 = d;  i2 = s; }
    }
    nn[((size_t)b * R + r) * 2 + 0] = i1;
    nn[((size_t)b * R + r) * 2 + 1] = i2;
  }
}

// ---------------------------------------------------------------- WMMA logsumexp kernel
// grid = (B, Q/QTILE), block = 128 threads (4 waves of 32; wave owns 32 q-rows).
// Per 16-col ref tile: one shared B fragment feeds two WMMAs (M-blocking),
// then per-lane streaming max/sumexp; final 16-lane shuffle combine ->
// logZ[b][q] = rowmax + log(sum exp).
__global__ void k_logz(const float* __restrict__ qry, const float* __restrict__ refm,
                       const float* __restrict__ ref_mask, const float* __restrict__ temperature,
                       float* __restrict__ logZ, int Q, int R, int D) {
  extern __shared__ float sm[];
  float* Qs = sm;                    // QTILE*D
  float* Rs = Qs + QTILE * D;        // 16*D
  float* Ms = Rs + 16 * D;           // ntiles*16
  const int b    = blockIdx.x;
  const int q0   = blockIdx.y * QTILE;
  const int tid  = threadIdx.x;
  const int lane = tid & 31;
  const int wave = tid >> 5;         // 0..3
  const int h    = lane >> 4;        // lane half
  const int nm   = lane & 15;        // row idx for A, col idx for B
  const int ntiles = (R + 1 + 15) / 16;     // 33 tiles covering cols 0..527
  const float invT = 1.0f / temperature[0];

  for (int i = tid; i < ntiles * 16; i += BLOCK)
    Ms[i] = (i < R) ? ref_mask[(size_t)b * R + i] : (i == R ? 1.0f : 0.0f);

  const float* qsrc = qry + ((size_t)b * Q + q0) * D;
  for (int i = tid * 4; i < QTILE * D; i += BLOCK * 4)
    *(float4*)(Qs + i) = *(const float4*)(qsrc + i);
  __syncthreads();

  float rm[16], rs[16];
#pragma unroll
  for (int j = 0; j < 16; ++j) { rm[j] = NEG_INF; rs[j] = 0.0f; }

  const float* rbase = refm + (size_t)b * (R + 1) * D;
  const float* arow0 = Qs + (wave * 32 + nm) * D;        // subtile 0 rows
  const float* arow1 = arow0 + 16 * D;                   // subtile 1 rows
  const float* brow  = Rs + nm * D;

  for (int t = 0; t < ntiles; ++t) {
    const int r0 = t * 16;
    // stage 16 ref rows (rows past R clamp to R; mask kills them)
    for (int i = tid * 4; i < 16 * D; i += BLOCK * 4) {
      int rr = i >> 8;              // D == 256
      int d  = i & (D - 1);
      int rsrc = r0 + rr; if (rsrc > R) rsrc = R;
      *(float4*)(Rs + i) = *(const float4*)(rbase + (size_t)rsrc * D + d);
      // pull next tile toward cache while this one computes
      int rnext = rsrc + 16; if (rnext > R) rnext = R;
      __builtin_prefetch(rbase + (size_t)rnext * D + d, 0, 1);
    }
    __syncthreads();

    v8f acc0 = {}, acc1 = {};
    for (int k = 0; k < D; k += 4) {
      v2f bb = *(const v2f*)(brow  + k + 2 * h);
      v2f a0 = *(const v2f*)(arow0 + k + 2 * h);
      v2f a1 = *(const v2f*)(arow1 + k + 2 * h);
      acc0 = __builtin_amdgcn_wmma_f32_16x16x4_f32(false, a0, false, bb,
                                                   (short)0, acc0, false, false);
      acc1 = __builtin_amdgcn_wmma_f32_16x16x4_f32(false, a1, false, bb,
                                                   (short)0, acc1, false, false);
    }

    float mval = Ms[r0 + nm];
    if (mval >= 0.5f) {
#pragma unroll
      for (int j = 0; j < 8; ++j) {
        float v   = acc0[j] * invT;
        float nmx = fmaxf(rm[j], v);
        rs[j] = rs[j] * __expf(rm[j] - nmx) + __expf(v - nmx);
        rm[j] = nmx;
        float v1   = acc1[j] * invT;
        float nmx1 = fmaxf(rm[j + 8], v1);
        rs[j + 8] = rs[j + 8] * __expf(rm[j + 8] - nmx1) + __expf(v1 - nmx1);
        rm[j + 8] = nmx1;
      }
    }
    __syncthreads();
  }

  // combine the 16 lanes of each half (xor stays inside the half)
#pragma unroll
  for (int sub = 0; sub < 2; ++sub) {
#pragma unroll
    for (int j = 0; j < 8; ++j) {
      float m = rm[sub * 8 + j], s = rs[sub * 8 + j];
      for (int off = 1; off < 16; off <<= 1) {
        float om = __shfl_xor(m, off, 32);
        float os = __shfl_xor(s, off, 32);
        float M  = fmaxf(m, om);
        s = s * __expf(m - M) + os * __expf(om - M);
        m = M;
      }
      if (nm == 0) {
        int qrow = q0 + wave * 32 + sub * 16 + j + 8 * h;
        logZ[(size_t)b * Q + qrow] = m + __logf(s);
      }
    }
  }
}

// ---------------------------------------------------------------- CE + hinge per (b,q)
__global__ void k_dots(const float* __restrict__ qry, const float* __restrict__ refm,
                       const float* __restrict__ ref_mask, const float* __restrict__ qry_mask,
                       const float* __restrict__ temperature, const int* __restrict__ match_idx,
                       const int* __restrict__ epoch_p, const int* __restrict__ nn,
                       const float* __restrict__ logZ, double* __restrict__ acc,
                       int B, int Q, int R, int D) {
  int idx = blockIdx.x * blockDim.x + threadIdx.x;
  if (idx >= B * Q) return;
  int b = idx / Q;
  float invT = 1.0f / temperature[0];
  int  y      = match_idx[idx];
  bool validq = qry_mask[idx] >= 0.5f;
  const float* qrow = qry + (size_t)idx * D;

  bool ymasked = (y < R) && (ref_mask[(size_t)b * R + y] < 0.5f);
  float simy;
  {
    const float* rrow = refm + ((size_t)b * (R + 1) + y) * D;
    float s = 0.0f;
    for (int d = 0; d < D; d += 4) {
      float4 a = *(const float4*)(qrow + d);
      float4 c = *(const float4*)(rrow + d);
      s += a.x * c.x + a.y * c.y + a.z * c.z + a.w * c.w;
    }
    simy = ymasked ? NEG_INF : s * invT;
  }

  double ce = 0.0, cec = 0.0, hs = 0.0, hc = 0.0;
  if (validq) { ce = (double)(logZ[idx] - simy); cec = 1.0; }

  int   epoch    = epoch_p[0];
  int   ramp_end = (int)(150 * 0.7);     // 105
  float t = fminf(1.0f, (float)epoch / (float)(ramp_end > 1 ? ramp_end : 1));
  float hard_w = t * 0.18f;
  int   hard_k = (int)roundf(t * 2.0f);
  if (hard_k < 0) hard_k = 0; if (hard_k > 2) hard_k = 2;

  if (hard_k > 0 && validq && y < R) {
    for (int i = 0; i < hard_k; ++i) {
      int j = nn[((size_t)b * R + y) * 2 + i];
      if (j < 0 || j == y) continue;
      const float* rrow = refm + ((size_t)b * (R + 1) + j) * D;
      float s = 0.0f;
      for (int d = 0; d < D; d += 4) {
        float4 a = *(const float4*)(qrow + d);
        float4 c = *(const float4*)(rrow + d);
        s += a.x * c.x + a.y * c.y + a.z * c.z + a.w * c.w;
      }
      float simj  = s * invT;
      float hinge = fmaxf(simj - simy + MARGIN_F, 0.0f);
      hs += (double)(hard_w * hinge);
      hc += 1.0;
    }
  }
  if (ce != 0.0 || cec != 0.0) { atomicAdd(&acc[0], ce); atomicAdd(&acc[1], cec); }
  if (hc != 0.0)               { atomicAdd(&acc[2], hs); atomicAdd(&acc[3], hc); }
}

// ---------------------------------------------------------------- temporal cosine term
__global__ void k_temporal(const float* __restrict__ qry, const float* __restrict__ refm,
                           const float* __restrict__ qry_mask, const float* __restrict__ ref_mask,
                           const int* __restrict__ pairs, double* __restrict__ acc,
                           int NP, int Q, int R, int D) {
  int i = blockIdx.x * blockDim.x + threadIdx.x;
  if (i >= NP) return;
  int tb = pairs[i * 3 + 0];
  int tq = pairs[i * 3 + 1];
  int tr = pairs[i * 3 + 2];
  const float* qv = qry  + ((size_t)tb * Q + tq) * D;
  const float* rv = refm + ((size_t)tb * (R + 1) + tr) * D;
  float dot = 0.f, nq = 0.f, nr = 0.f;
  for (int d = 0; d < D; d += 4) {
    float4 a = *(const float4*)(qv + d);
    float4 c = *(const float4*)(rv + d);
    dot += a.x * c.x + a.y * c.y + a.z * c.z + a.w * c.w;
    nq  += a.x * a.x + a.y * a.y + a.z * a.z + a.w * a.w;
    nr  += c.x * c.x + c.y * c.y + c.z * c.z + c.w * c.w;
  }
  float denom = fmaxf(sqrtf(nq) * sqrtf(nr), 1e-8f);
  float cs = dot / denom;
  bool valid = (qry_mask[(size_t)tb * Q + tq] >= 0.5f) &&
               (ref_mask[(size_t)tb * R + tr] >= 0.5f);
  if (valid) {
    atomicAdd(&acc[4], (double)(1.0f - cs));
    atomicAdd(&acc[5], 1.0);
  }
}

// ---------------------------------------------------------------- combine scalar
__global__ void k_final(const double* __restrict__ acc, float* __restrict__ out) {
  double ce = acc[0], cec = acc[1], hs = acc[2], hc = acc[3], ts = acc[4], tc = acc[5];
  double cnt   = cec + hc;
  double match = cnt > 0.0 ? (ce + hs) / fmax(cnt, 1.0) : 0.0;
  double tmean = tc  > 0.0 ? ts / fmax(tc, 1.0) : 0.0;
  out[0] = (float)(match + (double)TEMP_W * tmean);
}

// ---------------------------------------------------------------- host glue
extern "C" void kernel_launch(void* const* d_in, const int* in_sizes, int n_in,
                              void* d_out, int out_size, void* d_ws, size_t ws_size,
                              hipStream_t stream) {
  const float* ref_mean    = (const float*)d_in[0];
  const float* qry_mean    = (const float*)d_in[1];
  const float* temperature = (const float*)d_in[2];
  const float* ref_mask    = (const float*)d_in[3];
  const float* qry_mask    = (const float*)d_in[4];
  const float* coords      = (const float*)d_in[5];
  const int*   match_idx   = (const int*)d_in[6];
  const int*   pairs       = (const int*)d_in[7];
  const int*   epoch       = (const int*)d_in[8];

  long s0 = in_sizes[0];  // B*(R+1)*D
  long s1 = in_sizes[1];  // B*Q*D
  long s3 = in_sizes[3];  // B*R
  long s4 = in_sizes[4];  // B*Q
  int D  = (int)(s1 / s4);          // 256
  int B  = (int)(s0 / D - s3);      // s0/D = B*R + B
  int Q  = (int)(s4 / B);           // 512
  int R  = (int)(s3 / B);           // 512
  int NP = in_sizes[7] / 3;         // 4096

  double* acc  = (double*)d_ws;
  int*    nn   = (int*)((char*)d_ws + 64);
  float*  logZ = (float*)((char*)d_ws + 64 + (size_t)B * R * 2 * sizeof(int));

  k_init<<<1, 32, 0, stream>>>(acc);

  size_t nn_smem = (size_t)R * 4 * sizeof(float);
  k_nn<<<B, R, nn_smem, stream>>>(coords, ref_mask, nn, R);

  int ntiles = (R + 1 + 15) / 16;
  size_t logz_smem = ((size_t)QTILE * D + 16 * D + (size_t)ntiles * 16) * sizeof(float);
  (void)hipFuncSetAttribute((const void*)k_logz,
                            hipFuncAttributeMaxDynamicSharedMemorySize,
                            (int)logz_smem);
  dim3 gz(B, Q / QTILE);
  k_logz<<<gz, BLOCK, logz_smem, stream>>>(qry_mean, ref_mean, ref_mask, temperature,
                                           logZ, Q, R, D);

  int nq = B * Q;
  k_dots<<<(nq + 255) / 256, 256, 0, stream>>>(qry_mean, ref_mean, ref_mask, qry_mask,
                                               temperature, match_idx, epoch, nn, logZ,
                                               acc, B, Q, R, D);

  k_temporal<<<(NP + 255) / 256, 256, 0, stream>>>(qry_mean, ref_mean, qry_mask, ref_mask,
                                                   pairs, acc, NP, Q, R, D);

  k_final<<<1, 1, 0, stream>>>(acc, (float*)d_out);
}